// GenerateActionsBlock_22067541967154
// MI455X (gfx1250) — compile-verified
//
#include <hip/hip_runtime.h>

// GenerateActionsBlock fused kernel for MI455X (gfx1250, wave32).
// One workgroup per batch element; all 4 matmuls on V_WMMA_F32_16X16X4_F32.

#define E_DIM   128
#define N_ATOMS 128
#define BATCH   512
#define KX_ROW  257        // 2*E + 1
#define LSTRIDE 132        // padded LDS row stride (dwords) -> conflict-free b64 reads

typedef __attribute__((ext_vector_type(2))) float v2f;
typedef __attribute__((ext_vector_type(8))) float v8f;

__device__ __forceinline__ v8f wmma_f32(v2f a, v2f b, v8f c) {
  // 8 args: (neg_a, A, neg_b, B, c_mod, C, reuse_a, reuse_b)
  return __builtin_amdgcn_wmma_f32_16x16x4_f32(false, a, false, b, (short)0, c,
                                               false, false);
}

__launch_bounds__(256, 1)
__global__ void policy_fused_kernel(const float* __restrict__ kx,
                                    const float* __restrict__ positions,
                                    const float* __restrict__ atoms_mask,
                                    const float* __restrict__ action_scale,
                                    const float* __restrict__ eval_actions,
                                    const float* __restrict__ Wx,
                                    const float* __restrict__ bx,
                                    const float* __restrict__ Wk,
                                    const float* __restrict__ bk,
                                    float* __restrict__ out_actions,
                                    float* __restrict__ out_logprob) {
  __shared__ float sHx[N_ATOMS * LSTRIDE];   // 66 KB: H_x = softplus(x) @ Wx^T + bx
  __shared__ float sHk[N_ATOMS * LSTRIDE];   // 66 KB: H_k = relu(k)     @ Wk^T + bk
  __shared__ float sAx[16 * LSTRIDE];        // stage: activated x row-block
  __shared__ float sAk[16 * LSTRIDE];        // stage: activated k row-block
  __shared__ float sPos[N_ATOMS][3];
  __shared__ float sMask[N_ATOMS];
  __shared__ float sMean[N_ATOMS][3];
  __shared__ float sRed[8];

  const int b    = blockIdx.x;
  const int tid  = threadIdx.x;
  const int lane = tid & 31;
  const int wv   = tid >> 5;        // wave id 0..7
  const int lm   = lane & 15;       // lane mod 16
  const int hi   = lane >> 4;       // half-wave: selects K pair / M+8 rows

  const float INV_SQRT_E  = 0.08838834764831845f;  // 1/sqrt(128)
  const float LOG2F_C     = 0.6931471805599453f;
  const float PI_OVER_CUT = 0.6283185307179586f;   // pi / 5
  const float HALF_LOG2PI = 0.9189385332046727f;

  // ---- phase 0: positions + mask into LDS --------------------------------
  if (tid < N_ATOMS) {
    const size_t pb = ((size_t)b * N_ATOMS + tid) * 3;
    sPos[tid][0] = positions[pb + 0];
    sPos[tid][1] = positions[pb + 1];
    sPos[tid][2] = positions[pb + 2];
    sMask[tid]   = atoms_mask[(size_t)b * N_ATOMS + tid];
  }

  // ---- phase 1: per-wave weight B-fragments resident in VGPRs ------------
  // Wave wv owns output columns e in [16*wv, 16*wv+16). B-frag layout:
  // lane holds Wx[e0+lm][4s + 2*hi + {0,1}] for K-step s.
  v2f wxf[32], wkf[32];
  {
    const float* wxr = Wx + (size_t)(wv * 16 + lm) * E_DIM + 2 * hi;
    const float* wkr = Wk + (size_t)(wv * 16 + lm) * E_DIM + 2 * hi;
#pragma unroll
    for (int s = 0; s < 32; ++s) {
      wxf[s][0] = wxr[4 * s];  wxf[s][1] = wxr[4 * s + 1];
      wkf[s][0] = wkr[4 * s];  wkf[s][1] = wkr[4 * s + 1];
    }
  }
  const float bxv = bx[wv * 16 + lm];
  const float bkv = bk[wv * 16 + lm];

  // ---- phase 2: activations + GEMMs, H matrices into LDS -----------------
  for (int ib = 0; ib < 8; ++ib) {
    __syncthreads();  // previous iteration's stage reads done; sMask ready
    {
      const int   row = tid >> 4;            // 0..15
      const int   c0  = (tid & 15) * 8;      // 0..120
      const float m   = sMask[ib * 16 + row];
      const float* kxr = kx + ((size_t)b * N_ATOMS + ib * 16 + row) * KX_ROW;
#pragma unroll
      for (int u = 0; u < 8; ++u) {
        float xv = kxr[c0 + u] * m;
        // shifted softplus: log1p(exp(x)) - log(2), numerically stable
        float sp = fmaxf(xv, 0.0f) + log1pf(expf(-fabsf(xv))) - LOG2F_C;
        sAx[row * LSTRIDE + c0 + u] = sp;
        float kv = kxr[E_DIM + c0 + u] * m;
        sAk[row * LSTRIDE + c0 + u] = fmaxf(kv, 0.0f);
      }
    }
    __syncthreads();

    v8f accx, acck;
#pragma unroll
    for (int r = 0; r < 8; ++r) { accx[r] = bxv; acck[r] = bkv; }

    const float* axp = &sAx[lm * LSTRIDE + 2 * hi];
    const float* akp = &sAk[lm * LSTRIDE + 2 * hi];
#pragma unroll
    for (int s = 0; s < 32; ++s) {
      v2f a; a[0] = axp[4 * s]; a[1] = axp[4 * s + 1];
      accx = wmma_f32(a, wxf[s], accx);
    }
#pragma unroll
    for (int s = 0; s < 32; ++s) {
      v2f a; a[0] = akp[4 * s]; a[1] = akp[4 * s + 1];
      acck = wmma_f32(a, wkf[s], acck);
    }
    // C-tile layout: lane -> col e0+lm, VGPR r -> row r + 8*hi
#pragma unroll
    for (int r = 0; r < 8; ++r) {
      const int gr = ib * 16 + r + 8 * hi;
      sHx[gr * LSTRIDE + wv * 16 + lm] = accx[r];
      sHk[gr * LSTRIDE + wv * 16 + lm] = acck[r];
    }
  }
  __syncthreads();

  // ---- phase 3: Gram tiles + geometry fusion -----------------------------
  // Wave wv owns column block j in [16*wv, 16*wv+16). For H@H^T the B-frag
  // of rows j has the identical lane pattern as an A-frag -> preload once.
  v2f bxf[32], bkf[32];
  {
    const float* hxr = &sHx[(wv * 16 + lm) * LSTRIDE + 2 * hi];
    const float* hkr = &sHk[(wv * 16 + lm) * LSTRIDE + 2 * hi];
#pragma unroll
    for (int s = 0; s < 32; ++s) {
      bxf[s][0] = hxr[4 * s];  bxf[s][1] = hxr[4 * s + 1];
      bkf[s][0] = hkr[4 * s];  bkf[s][1] = hkr[4 * s + 1];
    }
  }
  const int   j   = wv * 16 + lm;
  const float pjx = sPos[j][0], pjy = sPos[j][1], pjz = sPos[j][2];
  const float mj  = sMask[j];
  float mx = 0.0f, my = 0.0f, mz = 0.0f;

  for (int ib = 0; ib < 8; ++ib) {
    v8f gx, gk;
#pragma unroll
    for (int r = 0; r < 8; ++r) { gx[r] = 0.0f; gk[r] = 0.0f; }

    const float* axr = &sHx[(ib * 16 + lm) * LSTRIDE + 2 * hi];
    const float* akr = &sHk[(ib * 16 + lm) * LSTRIDE + 2 * hi];
#pragma unroll
    for (int s = 0; s < 32; ++s) {
      v2f a; a[0] = axr[4 * s]; a[1] = axr[4 * s + 1];
      gx = wmma_f32(a, bxf[s], gx);
    }
#pragma unroll
    for (int s = 0; s < 32; ++s) {
      v2f a; a[0] = akr[4 * s]; a[1] = akr[4 * s + 1];
      gk = wmma_f32(a, bkf[s], gk);
    }

    // fuse: f = ((r_ij - Gx/sqrtE)*0.5) * (Gk/sqrtE); w = f * fcut * mi * mj
    // mean[j] += (pos_j - pos_i)/(r_ij + 1e-8) * w, summed over i
#pragma unroll
    for (int r = 0; r < 8; ++r) {
      const int   i   = ib * 16 + r + 8 * hi;
      const float dx  = pjx - sPos[i][0];
      const float dy  = pjy - sPos[i][1];
      const float dz  = pjz - sPos[i][2];
      const float rij = sqrtf(dx * dx + dy * dy + dz * dz + 1e-16f);
      const float inv = 1.0f / (rij + 1e-8f);
      const float xv  = (rij - gx[r] * INV_SQRT_E) * 0.5f;
      const float f   = xv * (gk[r] * INV_SQRT_E);
      const float fc  = (rij < 5.0f) ? 0.5f * (cosf(rij * PI_OVER_CUT) + 1.0f)
                                     : 0.0f;
      const float w   = f * fc * mj * sMask[i];
      const float sw  = w * inv;
      mx += dx * sw;  my += dy * sw;  mz += dz * sw;
    }
  }
  // combine the two half-wave row groups (rows r and r+8)
  mx += __shfl_xor(mx, 16, 32);
  my += __shfl_xor(my, 16, 32);
  mz += __shfl_xor(mz, 16, 32);
  if (lane < 16) { sMean[j][0] = mx; sMean[j][1] = my; sMean[j][2] = mz; }
  __syncthreads();

  // ---- phase 4: per-atom tail + log-prob reduction -----------------------
  float lp = 0.0f;
  const float as = action_scale[0];
  if (tid < N_ATOMS) {
    const int   n  = tid;
    const float m  = sMask[n];
    const float vx = sMean[n][0], vy = sMean[n][1], vz = sMean[n][2];
    const float nrm = sqrtf(vx * vx + vy * vy + vz * vz + 1e-16f) + 1e-8f;
    const float sc  = tanhf(nrm) / nrm * as;          // mean -> mu scaling
    const float mux = vx * sc, muy = vy * sc, muz = vz * sc;

    float ls = kx[((size_t)b * N_ATOMS + n) * KX_ROW + 2 * E_DIM] * m;
    ls = fminf(fmaxf(ls, -20.0f), 2.0f);
    const float sigma = expf(ls) * as;
    const float isig  = 1.0f / sigma;
    const float lsig  = logf(sigma);

    const size_t eb = ((size_t)b * N_ATOMS + n) * 3;
    const float a0 = eval_actions[eb + 0];
    const float a1 = eval_actions[eb + 1];
    const float a2 = eval_actions[eb + 2];
    out_actions[eb + 0] = a0 * m;
    out_actions[eb + 1] = a1 * m;
    out_actions[eb + 2] = a2 * m;

    const float z0 = (a0 - mux) * isig;
    const float z1 = (a1 - muy) * isig;
    const float z2 = (a2 - muz) * isig;
    lp = (-0.5f * (z0 * z0 + z1 * z1 + z2 * z2)
          - 3.0f * lsig - 3.0f * HALF_LOG2PI) * m;
  }
#pragma unroll
  for (int off = 16; off >= 1; off >>= 1) lp += __shfl_xor(lp, off, 32);
  if (lane == 0) sRed[wv] = lp;
  __syncthreads();
  if (tid == 0) {
    float t = 0.0f;
#pragma unroll
    for (int w = 0; w < 8; ++w) t += sRed[w];  // waves 4..7 contribute 0
    out_logprob[b] = t;
  }
}

extern "C" void kernel_launch(void* const* d_in, const int* in_sizes, int n_in,
                              void* d_out, int out_size, void* d_ws,
                              size_t ws_size, hipStream_t stream) {
  const float* kx  = (const float*)d_in[0];
  const float* pos = (const float*)d_in[1];
  const float* msk = (const float*)d_in[2];
  const float* asc = (const float*)d_in[3];
  const float* ea  = (const float*)d_in[4];
  const float* Wx  = (const float*)d_in[5];
  const float* bx  = (const float*)d_in[6];
  const float* Wk  = (const float*)d_in[7];
  const float* bk  = (const float*)d_in[8];

  float* out_actions = (float*)d_out;                         // B*N*3 floats
  float* out_logprob = out_actions + (size_t)BATCH * N_ATOMS * 3;  // B floats

  hipLaunchKernelGGL(policy_fused_kernel, dim3(BATCH), dim3(256), 0, stream,
                     kx, pos, msk, asc, ea, Wx, bx, Wk, bk,
                     out_actions, out_logprob);
}